// DFineMultiscaleDeformableAttention_22763326668843
// MI455X (gfx1250) — compile-verified
//
#include <hip/hip_runtime.h>
#include <hip/hip_bf16.h>

// Problem constants (fixed by the reference module)
#define NQ    500
#define DM    256
#define NH    8
#define HD    32
#define TP    12        // total sampling points per head
#define NOFF  192       // NH*TP*2  (W_off cols)
#define NATT  96        // NH*TP    (W_attn cols)
#define NTILES 18       // 12 offset col-tiles + 6 attn col-tiles (16 cols each)

#define LDA   260       // LDS stride for A tile (256 + 4 pad -> conflict-free column reads)
#define LDO   292       // LDS stride for output tile (288 + 4 pad)

typedef __attribute__((ext_vector_type(2))) float v2f;
typedef __attribute__((ext_vector_type(8))) float v8f;

// ---------------------------------------------------------------------------
// Kernel 1: fused projection GEMM (fp32 WMMA) + bias + softmax + location math
//   grid.x = M/16 blocks, 288 threads = 9 waves. Wave w statically owns
//   N-tiles {2w, 2w+1} -> two named v8f accumulators, zero dynamic indexing,
//   so the accumulators stay in VGPRs and every K-step issues 2 WMMAs.
// ---------------------------------------------------------------------------
__global__ __launch_bounds__(288) void dfine_proj_kernel(
    const float* __restrict__ hidden,  // [M, 256]
    const float* __restrict__ refpts,  // [M, 4]
    const float* __restrict__ Woff,    // [256, 192]
    const float* __restrict__ boff,    // [192]
    const float* __restrict__ Watt,    // [256, 96]
    const float* __restrict__ batt,    // [96]
    float* __restrict__ params,        // [M*8, 36]  (12 attn | 12 px | 12 py)
    int M)
{
    __shared__ float sA[16 * LDA];     // 16 rows x 256 K, fp32
    __shared__ float sO[16 * LDO];     // 16 rows x 288 cols of D

    const int tid  = threadIdx.x;
    const int wave = tid >> 5;         // 0..8
    const int lane = tid & 31;
    const int m0   = blockIdx.x * 16;

    // --- stage A tile (coalesced: consecutive tid -> consecutive K) ---
    for (int idx = tid; idx < 16 * DM; idx += 288) {
        int row = idx >> 8;            // / 256
        int col = idx & 255;
        int m   = m0 + row;
        if (m >= M) m = M - 1;         // clamp (8000 is a multiple of 16 anyway)
        sA[row * LDA + col] = hidden[(size_t)m * DM + col];
    }
    __syncthreads();

    const int halfsel = lane >> 4;     // 0: lanes 0-15 (K=0,1) ; 1: lanes 16-31 (K=2,3)
    const int l16     = lane & 15;

    // Static tile pair for this wave (wave-uniform selection, done once)
    const int t0 = wave * 2;
    const int t1 = t0 + 1;

    const float* B0; int ldb0;
    if (t0 < 12) { B0 = Woff + t0 * 16;        ldb0 = NOFF; }
    else         { B0 = Watt + (t0 - 12) * 16; ldb0 = NATT; }
    const float* B1; int ldb1;
    if (t1 < 12) { B1 = Woff + t1 * 16;        ldb1 = NOFF; }
    else         { B1 = Watt + (t1 - 12) * 16; ldb1 = NATT; }

    const float* b0p = B0 + l16;       // per-lane column pointer
    const float* b1p = B1 + l16;
    const float* arp = sA + l16 * LDA; // per-lane A row pointer (LDS)

    v8f acc0 = {};
    v8f acc1 = {};

#pragma unroll 4
    for (int k = 0; k < DM; k += 4) {
        const int ka = k + 2 * halfsel;
        // A fragment: A[M=l16][K=ka..ka+1]  (16x4 fp32 layout, ISA 7.12.2)
        v2f a;
        a.x = arp[ka + 0];
        a.y = arp[ka + 1];
        // B fragments: B[K=ka..ka+1][N=l16]
        v2f b0;
        b0.x = b0p[(size_t)(ka + 0) * ldb0];
        b0.y = b0p[(size_t)(ka + 1) * ldb0];
        v2f b1;
        b1.x = b1p[(size_t)(ka + 0) * ldb1];
        b1.y = b1p[(size_t)(ka + 1) * ldb1];
        acc0 = __builtin_amdgcn_wmma_f32_16x16x4_f32(
            false, a, false, b0, (short)0, acc0, false, false);
        acc1 = __builtin_amdgcn_wmma_f32_16x16x4_f32(
            false, a, false, b1, (short)0, acc1, false, false);
    }

    // --- spill D tiles to LDS: d[r] -> row = r + 8*halfsel, col = t*16 + l16 ---
    {
        const int so0 = t0 * 16 + l16;
        const int so1 = t1 * 16 + l16;
#pragma unroll
        for (int r = 0; r < 8; ++r) {
            const int row = r + 8 * halfsel;
            sO[row * LDO + so0] = acc0[r];
            sO[row * LDO + so1] = acc1[r];
        }
    }
    __syncthreads();

    // --- epilogue: one thread per (row, head): bias + softmax(12) + locations ---
    if (tid < 128) {
        const int row  = tid >> 3;
        const int head = tid & 7;
        const int m    = m0 + row;
        if (m < M) {
            const float rx = refpts[m * 4 + 0];
            const float ry = refpts[m * 4 + 1];
            const float rw = refpts[m * 4 + 2];
            const float rh = refpts[m * 4 + 3];

            float logit[TP], lx[TP], ly[TP];
            float mx = -1e30f;
#pragma unroll
            for (int p = 0; p < TP; ++p) {
                const int co = head * 24 + p * 2;
                const float ox = sO[row * LDO + co + 0] + boff[co + 0];
                const float oy = sO[row * LDO + co + 1] + boff[co + 1];
                const int   ca = head * 12 + p;
                const float lg = sO[row * LDO + NOFF + ca] + batt[ca];
                // loc = ref_xy + off * (1/4) * 0.5 * ref_wh  -> 0.125
                const float locx = rx + ox * 0.125f * rw;
                const float locy = ry + oy * 0.125f * rh;
                const int   sz   = 80 >> (p >> 2);          // level 0/1/2 -> 80/40/20
                lx[p] = locx * (float)sz - 0.5f;
                ly[p] = locy * (float)sz - 0.5f;
                logit[p] = lg;
                mx = fmaxf(mx, lg);
            }
            float s = 0.f;
#pragma unroll
            for (int p = 0; p < TP; ++p) { logit[p] = __expf(logit[p] - mx); s += logit[p]; }
            const float inv = 1.0f / s;

            float* o = params + ((size_t)m * NH + head) * 36;
#pragma unroll
            for (int p = 0; p < TP; ++p) {
                o[p]      = logit[p] * inv;
                o[12 + p] = lx[p];
                o[24 + p] = ly[p];
            }
        }
    }
}

// ---------------------------------------------------------------------------
// Kernel 2: bilinear gather + weighted accumulation.
//   One wave32 per (m, head); lane = channel. px/py/attn are wave-uniform, so
//   every corner fetch is a single coalesced 128B global load; predicates for
//   zero-padding are uniform (no lane divergence).
// ---------------------------------------------------------------------------
__global__ __launch_bounds__(256) void dfine_sample_kernel(
    const float* __restrict__ value,   // [bs, 8400, 256]
    const float* __restrict__ params,  // [M*8, 36]
    float* __restrict__ out,           // [M, 256]
    int M)
{
    const int lane = threadIdx.x & 31;
    const int task = blockIdx.x * 8 + (threadIdx.x >> 5);
    if (task >= M * NH) return;
    const int head = task & 7;
    const int m    = task >> 3;
    const int b    = m / NQ;

    const float* pp    = params + (size_t)task * 36;
    const float* vbase = value + (size_t)b * 8400 * DM + head * HD + lane;

    float acc = 0.f;
#pragma unroll
    for (int p = 0; p < TP; ++p) {
        const int lvl   = p >> 2;
        const int sz    = 80 >> lvl;                               // H == W
        const int tbase = (lvl == 0) ? 0 : (lvl == 1 ? 6400 : 8000);

        const float a  = pp[p];
        const float px = pp[12 + p];
        const float py = pp[24 + p];

        const float x0f = floorf(px), y0f = floorf(py);
        const float wx = px - x0f, wy = py - y0f;
        const int x0 = (int)x0f, y0 = (int)y0f;

        float v00 = 0.f, v01 = 0.f, v10 = 0.f, v11 = 0.f;
        const bool xin0 = (x0 >= 0) && (x0 < sz);
        const bool xin1 = (x0 + 1 >= 0) && (x0 + 1 < sz);
        if ((y0 >= 0) && (y0 < sz)) {
            const float* rowp = vbase + (size_t)(tbase + y0 * sz) * DM;
            if (xin0) v00 = rowp[(size_t)x0 * DM];
            if (xin1) v01 = rowp[(size_t)(x0 + 1) * DM];
        }
        if ((y0 + 1 >= 0) && (y0 + 1 < sz)) {
            const float* rowp = vbase + (size_t)(tbase + (y0 + 1) * sz) * DM;
            if (xin0) v10 = rowp[(size_t)x0 * DM];
            if (xin1) v11 = rowp[(size_t)(x0 + 1) * DM];
        }
        const float top = v00 + wx * (v01 - v00);
        const float bot = v10 + wx * (v11 - v10);
        acc += a * (top + wy * (bot - top));
    }
    out[(size_t)m * DM + head * HD + lane] = acc;
}

// ---------------------------------------------------------------------------
extern "C" void kernel_launch(void* const* d_in, const int* in_sizes, int n_in,
                              void* d_out, int out_size, void* d_ws, size_t ws_size,
                              hipStream_t stream)
{
    const float* hidden = (const float*)d_in[0];   // [bs, nq, 256]
    const float* enc    = (const float*)d_in[1];   // [bs, 8400, 256]
    const float* refp   = (const float*)d_in[2];   // [bs, nq, 4]
    const float* Woff   = (const float*)d_in[3];   // [256, 192]
    const float* boff   = (const float*)d_in[4];   // [192]
    const float* Watt   = (const float*)d_in[5];   // [256, 96]
    const float* batt   = (const float*)d_in[6];   // [96]
    float* out    = (float*)d_out;
    float* params = (float*)d_ws;                  // [M*8, 36] fp32 (~9 MB)

    const int M = in_sizes[0] / DM;                // bs * nq = 8000

    const int mblocks = (M + 15) / 16;
    dfine_proj_kernel<<<mblocks, 288, 0, stream>>>(hidden, refp, Woff, boff,
                                                   Watt, batt, params, M);

    const int tasks   = M * NH;
    const int sblocks = (tasks + 7) / 8;
    dfine_sample_kernel<<<sblocks, 256, 0, stream>>>(enc, params, out, M);
}